// GraphConvolutionalBlock_10213432229958
// MI455X (gfx1250) — compile-verified
//
#include <hip/hip_runtime.h>

// ---------------------------------------------------------------------------
// GraphConv stack on MI455X (gfx1250): fp32 WMMA GEMMs + atomic edge scatter.
// ---------------------------------------------------------------------------

typedef __attribute__((ext_vector_type(2))) float v2f;
typedef __attribute__((ext_vector_type(8))) float v8f;

#define NNODES   50000
#define NEDGES   400000
#define DIN      512
#define DOUT     256
#define NHIDDEN  12

// ---------------------------------------------------------------------------
// GEMM: out[m, n] = sum_k X[m,k] * W[k,n] + bias[n]
//   X: [NNODES, K] row-major, W: [K, DOUT] row-major, out: [NNODES, DOUT]
// Block = 256 threads = 8 wave32. Each block computes a 16-row strip across
// all 256 output columns; each wave owns two 16x16 N-tiles.
// A strip (16 contiguous rows) is staged in LDS and reused by all 8 waves.
//
// Fragment layouts per CDNA5 ISA 7.12.2 (wave32, V_WMMA_F32_16X16X4_F32):
//   A (16x4):  lane&15 = M row, lane>>4 selects K pair {0,1} vs {2,3};
//              VGPR j holds K = (lane>>4)*2 + j.
//   B (4x16):  lane&15 = N col, lane>>4 selects K pair; VGPR j holds K pair+j.
//   C/D(16x16):lane&15 = N col, VGPR v holds M = v + (lane>>4)*8.
// ---------------------------------------------------------------------------
template <int K>
__global__ __launch_bounds__(256, 2) void gemm16_bias_wmma(
    const float* __restrict__ X, const float* __restrict__ W,
    const float* __restrict__ bias, float* __restrict__ out) {
  __shared__ float As[16 * K];

  const int tid = threadIdx.x;
  const size_t m0 = (size_t)blockIdx.x * 16;

  // Stage the 16xK A strip: 16 consecutive rows are one contiguous region.
  {
    const float4* __restrict__ src =
        reinterpret_cast<const float4*>(X + m0 * K);
    float4* dst = reinterpret_cast<float4*>(As);
#pragma unroll
    for (int i = 0; i < (16 * K) / (4 * 256); ++i)
      dst[tid + i * 256] = src[tid + i * 256];
  }
  __syncthreads();

  const int lane = tid & 31;
  const int wave = tid >> 5;
  const int r16 = lane & 15;  // A-frag row / B,C-frag column
  const int kh = lane >> 4;   // which K-half of the 4-wide K slab
  const int n0 = wave * 32;   // this wave's 32 output columns (two tiles)

  v8f acc0 = {0.f, 0.f, 0.f, 0.f, 0.f, 0.f, 0.f, 0.f};
  v8f acc1 = {0.f, 0.f, 0.f, 0.f, 0.f, 0.f, 0.f, 0.f};

  const float* __restrict__ Arow = As + r16 * K + kh * 2;

#pragma unroll 4
  for (int k = 0; k < K; k += 4) {
    v2f a;
    a.x = Arow[k];      // contiguous pair -> ds_load_b64
    a.y = Arow[k + 1];

    const float* __restrict__ w =
        W + (size_t)(k + kh * 2) * DOUT + n0 + r16;
    v2f b0, b1;
    b0.x = w[0];
    b0.y = w[DOUT];
    b1.x = w[16];
    b1.y = w[DOUT + 16];

    acc0 = __builtin_amdgcn_wmma_f32_16x16x4_f32(false, a, false, b0,
                                                 (short)0, acc0, false, false);
    acc1 = __builtin_amdgcn_wmma_f32_16x16x4_f32(false, a, false, b1,
                                                 (short)0, acc1, false, false);
  }

  const float bc0 = bias[n0 + r16];
  const float bc1 = bias[n0 + 16 + r16];
#pragma unroll
  for (int v = 0; v < 8; ++v) {
    const size_t row = m0 + (size_t)(v + kh * 8);
    out[row * DOUT + n0 + r16] = acc0[v] + bc0;
    out[row * DOUT + n0 + 16 + r16] = acc1[v] + bc1;
  }
}

// ---------------------------------------------------------------------------
// Edge scatter: for edge (a,b): h0[a,:] += h1[b,:]; h0[b,:] += h1[a,:]
// One block (256 lanes) per edge; coalesced h1 reads, coalesced f32 atomics.
// ---------------------------------------------------------------------------
__global__ __launch_bounds__(256) void edge_scatter(
    float* __restrict__ h0, const float* __restrict__ h1,
    const int* __restrict__ edges) {
  const int e = blockIdx.x;
  const int t = threadIdx.x;
  const int a = edges[2 * e];
  const int b = edges[2 * e + 1];
  const float va = h1[(size_t)b * DOUT + t];
  const float vb = h1[(size_t)a * DOUT + t];
  atomicAdd(&h0[(size_t)a * DOUT + t], va);
  atomicAdd(&h0[(size_t)b * DOUT + t], vb);
}

// ---------------------------------------------------------------------------
// Elementwise: x = relu(x) in place; optional residual snapshot.
// ---------------------------------------------------------------------------
__global__ __launch_bounds__(256) void relu_save(float* __restrict__ x,
                                                 float* __restrict__ res,
                                                 int n4) {
  const int i = blockIdx.x * 256 + threadIdx.x;
  if (i >= n4) return;
  float4 v = reinterpret_cast<float4*>(x)[i];
  v.x = fmaxf(v.x, 0.f);
  v.y = fmaxf(v.y, 0.f);
  v.z = fmaxf(v.z, 0.f);
  v.w = fmaxf(v.w, 0.f);
  reinterpret_cast<float4*>(x)[i] = v;
  if (res) reinterpret_cast<float4*>(res)[i] = v;
}

__global__ __launch_bounds__(256) void add_inplace(float* __restrict__ x,
                                                   const float* __restrict__ r,
                                                   int n4) {
  const int i = blockIdx.x * 256 + threadIdx.x;
  if (i >= n4) return;
  float4 v = reinterpret_cast<float4*>(x)[i];
  const float4 u = reinterpret_cast<const float4*>(r)[i];
  v.x += u.x;
  v.y += u.y;
  v.z += u.z;
  v.w += u.w;
  reinterpret_cast<float4*>(x)[i] = v;
}

// ---------------------------------------------------------------------------
extern "C" void kernel_launch(void* const* d_in, const int* in_sizes, int n_in,
                              void* d_out, int out_size, void* d_ws,
                              size_t ws_size, hipStream_t stream) {
  const float* features = (const float*)d_in[0];
  const int* edges = (const int*)d_in[1];
  const float* w0_in = (const float*)d_in[2];
  const float* b0_in = (const float*)d_in[3];
  const float* w1_in = (const float*)d_in[4];
  const float* b1_in = (const float*)d_in[5];
  const float* w0_h = (const float*)d_in[6];
  const float* b0_h = (const float*)d_in[7];
  const float* w1_h = (const float*)d_in[8];
  const float* b1_h = (const float*)d_in[9];
  float* out = (float*)d_out;

  const size_t NB = (size_t)NNODES * DOUT;  // 12.8M floats per buffer
  float* bufA = (float*)d_ws;
  float* bufB = bufA + NB;
  float* h1 = bufB + NB;
  float* res = h1 + NB;

  const int mtiles = NNODES / 16;          // 3125
  const int n4 = (int)(NB / 4);            // float4 count
  const int egrid = NEDGES;
  const int ew = (n4 + 255) / 256;

  // ---- Layer 0: DIN -> DOUT, save residual --------------------------------
  gemm16_bias_wmma<DIN><<<mtiles, 256, 0, stream>>>(features, w0_in, b0_in,
                                                    bufA);
  gemm16_bias_wmma<DIN><<<mtiles, 256, 0, stream>>>(features, w1_in, b1_in,
                                                    h1);
  edge_scatter<<<egrid, 256, 0, stream>>>(bufA, h1, edges);
  relu_save<<<ew, 256, 0, stream>>>(bufA, res, n4);

  // ---- 12 hidden layers: DOUT -> DOUT -------------------------------------
  float* x = bufA;
  float* y = bufB;
  for (int i = 0; i < NHIDDEN; ++i) {
    const float* w0 = w0_h + (size_t)i * DOUT * DOUT;
    const float* w1 = w1_h + (size_t)i * DOUT * DOUT;
    const float* bb0 = b0_h + (size_t)i * DOUT;
    const float* bb1 = b1_h + (size_t)i * DOUT;
    gemm16_bias_wmma<DOUT><<<mtiles, 256, 0, stream>>>(x, w0, bb0, y);
    gemm16_bias_wmma<DOUT><<<mtiles, 256, 0, stream>>>(x, w1, bb1, h1);
    edge_scatter<<<egrid, 256, 0, stream>>>(y, h1, edges);
    relu_save<<<ew, 256, 0, stream>>>(y, nullptr, n4);
    float* t = x;
    x = y;
    y = t;
  }

  // ---- Residual add, final layer into d_out -------------------------------
  add_inplace<<<ew, 256, 0, stream>>>(x, res, n4);

  const float* w0f = w0_h + (size_t)NHIDDEN * DOUT * DOUT;
  const float* w1f = w1_h + (size_t)NHIDDEN * DOUT * DOUT;
  const float* b0f = b0_h + (size_t)NHIDDEN * DOUT;
  const float* b1f = b1_h + (size_t)NHIDDEN * DOUT;
  gemm16_bias_wmma<DOUT><<<mtiles, 256, 0, stream>>>(x, w0f, b0f, out);
  gemm16_bias_wmma<DOUT><<<mtiles, 256, 0, stream>>>(x, w1f, b1f, h1);
  edge_scatter<<<egrid, 256, 0, stream>>>(out, h1, edges);
  relu_save<<<ew, 256, 0, stream>>>(out, nullptr, n4);
}